// CylinderQueryAndGroup_19121194402077
// MI455X (gfx1250) — compile-verified
//
#include <hip/hip_runtime.h>
#include <hip/hip_bf16.h>

// ---------------------------------------------------------------------------
// CylinderQueryAndGroup for MI455X (gfx1250, wave32)
//
// Shapes (fixed by reference setup_inputs): B=2, N=16384, P=1024, C=256, S=64
// Memory-bound op (~170 MB HBM traffic -> ~7.3us floor @ 23.3 TB/s; features
// tensor (34MB) and output (136MB) dominate; xyz is L2-resident).
//
// K1: per-(b,p) cylinder scores resident in LDS; xyz swept through LDS with
//     double-buffered GLOBAL_LOAD_ASYNC_TO_LDS_B128 (ASYNCcnt-tracked, no
//     VGPR round-trip); iterative argmin top-64 with smallest-index
//     tie-break + d2 fallback.
// K2: coalesced feature gather (idx staged in LDS).
// K3: grouped-xyz @ rot via V_WMMA_F32_16X16X4_F32; accumulator layout gives
//     each active lane 8 consecutive samples -> contiguous b128 stores
//     (confirmed in round-1 disasm).
// ---------------------------------------------------------------------------

#define B_ 2
#define N_ 16384
#define P_ 1024
#define C_ 256
#define S_ 64
#define CH_ (3 + C_)   // 259 output channels

#define RADIUS_ 0.05f
#define HMIN_ (-0.02f)
#define HMAX_ 0.04f

// xyz staging tiles: 2048 points * 12B = 24KB per tile, 8 tiles per batch.
#define TILE_PTS_ 2048
#define TILE_BYTES_ (TILE_PTS_ * 12)
#define N_TILES_ (N_ / TILE_PTS_)
#define CHUNKS_PER_THREAD_ (TILE_BYTES_ / 16 / 256)   // 6 x b128 per thread

typedef __attribute__((ext_vector_type(2))) float v2f;
typedef __attribute__((ext_vector_type(8))) float v8f;

#if defined(__has_builtin)
#if __has_builtin(__builtin_amdgcn_global_load_async_to_lds_b128) && \
    __has_builtin(__builtin_amdgcn_s_wait_asynccnt)
#define USE_ASYNC_LDS 1
#endif
#endif
#ifndef USE_ASYNC_LDS
#define USE_ASYNC_LDS 0
#endif

#if USE_ASYNC_LDS
// b128 payload type exactly as the builtin expects it:
//   '__attribute__((__vector_size__(4 * sizeof(int)))) int __device__ *'
// (HIP prints addrspace(1) as __device__, addrspace(3) as __shared__).
typedef int b128_t __attribute__((vector_size(4 * sizeof(int))));
typedef __attribute__((address_space(1))) b128_t as1_b128;
typedef __attribute__((address_space(3))) b128_t as3_b128;
#endif

// Issue one 24KB tile copy: global xyz -> LDS buffer. 256 threads x 6 b128.
__device__ __forceinline__ void stage_tile(const float* __restrict__ xb,
                                           int tile, char* lds_buf, int tid) {
  const char* g = (const char*)xb + (size_t)tile * TILE_BYTES_;
#if USE_ASYNC_LDS
#pragma unroll
  for (int j = 0; j < CHUNKS_PER_THREAD_; ++j) {
    const int off = (j * 256 + tid) * 16;
    __builtin_amdgcn_global_load_async_to_lds_b128(
        (as1_b128*)(g + off), (as3_b128*)(lds_buf + off),
        /*imm offset=*/0, /*cpol=*/0);
  }
#else
#pragma unroll
  for (int j = 0; j < CHUNKS_PER_THREAD_; ++j) {
    const int off = (j * 256 + tid) * 16;
    *(float4*)(lds_buf + off) = *(const float4*)(g + off);
  }
#endif
}

__device__ __forceinline__ void wait_tile_inflight() {
#if USE_ASYNC_LDS
  // one tile (6 ops/thread) may remain outstanding; older tile has landed
  // (async loads complete in order)
  __builtin_amdgcn_s_wait_asynccnt(CHUNKS_PER_THREAD_);
#endif
}
__device__ __forceinline__ void wait_tile_all() {
#if USE_ASYNC_LDS
  __builtin_amdgcn_s_wait_asynccnt(0);
#endif
}

// (value,index) argmin tree over 256 threads; smaller index wins ties
// (matches jax.lax.top_k tie order). Inputs pre-stored in rv/ri by caller.
__device__ __forceinline__ void argmin_reduce256(float* rv, int* ri, int tid) {
  for (int off = 128; off > 0; off >>= 1) {
    __syncthreads();
    if (tid < off) {
      float v2 = rv[tid + off];
      int   i2 = ri[tid + off];
      float v1 = rv[tid];
      int   i1 = ri[tid];
      if (v2 < v1 || (v2 == v1 && i2 < i1)) { rv[tid] = v2; ri[tid] = i2; }
    }
  }
  __syncthreads();
}

// --------------------------- Kernel 1: query -------------------------------
// One 256-thread workgroup per (b,p). Dynamic LDS layout (16B-aligned parts):
//   [0, 64KB)            : float score[N]
//   [64KB, 64KB+48KB)    : 2 x 24KB xyz staging buffers
//   then                 : float rv[256]; int ri[256]; int s_idx[64]; int misc[4]
__global__ void cyl_query_kernel(const float* __restrict__ xyz,
                                 const float* __restrict__ new_xyz,
                                 const float* __restrict__ rot,
                                 int* __restrict__ ws_idx) {
  extern __shared__ char smem[];
  float* s_score = (float*)smem;
  char*  s_buf0  = smem + (size_t)N_ * 4;
  char*  s_buf1  = s_buf0 + TILE_BYTES_;
  float* rv      = (float*)(s_buf1 + TILE_BYTES_);
  int*   ri      = (int*)((char*)rv + 256 * 4);
  int*   s_idx   = (int*)((char*)ri + 256 * 4);
  int*   s_misc  = (int*)((char*)s_idx + S_ * 4);

  const int tid = threadIdx.x;
  const int bp  = blockIdx.x;      // b*P + p
  const int b   = bp / P_;

  const float cx = new_xyz[bp * 3 + 0];
  const float cy = new_xyz[bp * 3 + 1];
  const float cz = new_xyz[bp * 3 + 2];
  const float* R = rot + (size_t)bp * 9;
  const float r00 = R[0], r01 = R[1], r02 = R[2];
  const float r10 = R[3], r11 = R[4], r12 = R[5];
  const float r20 = R[6], r21 = R[7], r22 = R[8];

  const float INF = __builtin_inff();
  const float* xb = xyz + (size_t)b * N_ * 3;

  // Pass 1: double-buffered async staging of xyz tiles -> score all N points.
  float best_d2 = INF;
  int   best_n  = 0;

  stage_tile(xb, 0, s_buf0, tid);
  for (int t = 0; t < N_TILES_; ++t) {
    char* cur = (t & 1) ? s_buf1 : s_buf0;
    if (t + 1 < N_TILES_) {
      stage_tile(xb, t + 1, ((t + 1) & 1) ? s_buf1 : s_buf0, tid);
      wait_tile_inflight();     // tile t landed; tile t+1 may be in flight
    } else {
      wait_tile_all();
    }
    __syncthreads();            // all waves' portions of tile t visible

    const float* pts = (const float*)cur;
    for (int i = tid; i < TILE_PTS_; i += 256) {
      const int n = t * TILE_PTS_ + i;
      float px = pts[i * 3 + 0];     // stride-3 floats: bank-conflict-free
      float py = pts[i * 3 + 1];
      float pz = pts[i * 3 + 2];
      float dx = px - cx, dy = py - cy, dz = pz - cz;
      float u = dx * r00 + dy * r01 + dz * r02;   // cylinder axis coord
      float v = dx * r10 + dy * r11 + dz * r12;
      float w = dx * r20 + dy * r21 + dz * r22;
      float rad = sqrtf(fmaxf(v * v + w * w, 0.0f));
      bool  in  = (rad <= RADIUS_) && (u >= HMIN_) && (u <= HMAX_);
      s_score[n] = in ? (rad + 1e-3f * fabsf(u)) : INF;
      float d2 = dx * dx + dy * dy + dz * dz;
      if (d2 < best_d2) { best_d2 = d2; best_n = n; }
    }
    __syncthreads();            // tile t fully consumed before its buffer
  }                             // is re-targeted by tile t+2's issue

  rv[tid] = best_d2;
  ri[tid] = best_n;
  argmin_reduce256(rv, ri, tid);
  if (tid == 0) s_misc[0] = ri[0];   // fallback = argmin d2 (first occurrence)
  __syncthreads();

  // Pass 2: 64 x argmin-with-removal. Removed winners marked NaN (never
  // selected again: NaN loses every comparison below).
  for (int k = 0; k < S_; ++k) {
    float bv = INF;
    int   bi = 0x7FFFFFFF;
    for (int n = tid; n < N_; n += 256) {
      float v = s_score[n];
      if (v < bv || (v == bv && n < bi)) { bv = v; bi = n; }
    }
    rv[tid] = bv;
    ri[tid] = bi;
    argmin_reduce256(rv, ri, tid);
    if (tid == 0) {
      int win = ri[0];
      if (k == 0) s_misc[1] = (rv[0] == INF) ? 1 : 0;  // no point in cylinder
      if (win == 0x7FFFFFFF) win = 0;                   // safety, unreachable
      s_idx[k] = win;
      s_score[win] = __builtin_nanf("");
    }
    __syncthreads();
  }

  if (tid < S_) {
    int outv = s_misc[1] ? s_misc[0] : s_idx[tid];
    ws_idx[(size_t)bp * S_ + tid] = outv;
  }
}

// ----------------------- Kernel 2: feature gather --------------------------
// block = 64 samples x 4 channels; grid = (B*P, C/4).
// Writes are coalesced along S; reads gather L2-resident feature rows.
__global__ void group_feat_kernel(const float* __restrict__ features,
                                  const int* __restrict__ ws_idx,
                                  float* __restrict__ out) {
  __shared__ int s_idx[S_];
  const int tid = threadIdx.x;
  const int bp  = blockIdx.x;
  const int b   = bp / P_;
  const int p   = bp % P_;
  if (tid < S_) s_idx[tid] = ws_idx[(size_t)bp * S_ + tid];
  __syncthreads();
  const int s = tid & (S_ - 1);
  const int c = blockIdx.y * 4 + (tid >> 6);
  const int n = s_idx[s];
  const float val = features[((size_t)b * C_ + c) * N_ + n];
  out[(((size_t)b * CH_ + 3 + c) * P_ + p) * S_ + s] = val;
}

// ------------------- Kernel 3: rotate grouped xyz (WMMA) -------------------
// One wave (32 threads) per (b,p). D = A(16x4) x B(4x16):
//   A[m,k] = delta_k of sample m  (lanes 0-15: K=0,1 = dx,dy; lanes 16-31:
//            K=2,3 = dz,0  -- per CDNA5 32-bit A 16x4 VGPR layout)
//   B[k,n] = rot[k][n] for n<3 else 0  (rows striped across lanes)
// D[m,n] = sum_d delta_d * rot[d][n]  == grouped_xyz @ rot.
// Accumulator layout: lane L holds column N=L%16, rows M=v+8*(L>=16) for
// VGPR v -> each active lane stores 8 consecutive samples (b128 pairs).
__global__ void rotate_xyz_wmma_kernel(const float* __restrict__ xyz,
                                       const float* __restrict__ new_xyz,
                                       const float* __restrict__ rot,
                                       const int* __restrict__ ws_idx,
                                       float* __restrict__ out) {
  const int lane = threadIdx.x;     // 0..31 (wave32)
  const int bp   = blockIdx.x;
  const int b    = bp / P_;
  const int p    = bp % P_;
  const int ncol = lane & 15;
  const bool hi  = lane >= 16;

  const float cx = new_xyz[bp * 3 + 0];
  const float cy = new_xyz[bp * 3 + 1];
  const float cz = new_xyz[bp * 3 + 2];
  const float* R = rot + (size_t)bp * 9;

  // B matrix: columns 0..2 hold rot rows; rest zero. EXEC reconverges before
  // each WMMA (all-1s requirement).
  float b0 = (ncol < 3) ? (hi ? R[6 + ncol] : R[ncol]) : 0.0f;
  float b1 = (ncol < 3 && !hi) ? R[3 + ncol] : 0.0f;
  v2f Bm = {b0, b1};

  const int*   idx = ws_idx + (size_t)bp * S_;
  const float* xb  = xyz + (size_t)b * N_ * 3;

#pragma unroll
  for (int t = 0; t < 4; ++t) {           // 4 tiles of 16 samples = 64
    const int s = t * 16 + ncol;
    const int n = idx[s];
    const float* pt = xb + (size_t)n * 3;
    float a0 = hi ? (pt[2] - cz) : (pt[0] - cx);
    float a1 = hi ? 0.0f : (pt[1] - cy);
    v2f Am = {a0, a1};
    v8f acc = {};
    acc = __builtin_amdgcn_wmma_f32_16x16x4_f32(
        /*neg_a=*/false, Am, /*neg_b=*/false, Bm,
        /*c_mod=*/(short)0, acc, /*reuse_a=*/false, /*reuse_b=*/false);
    if (ncol < 3) {                        // lanes {0,1,2,16,17,18} store
      float* o = out + (((size_t)b * CH_ + ncol) * P_ + p) * S_
                     + t * 16 + (hi ? 8 : 0);
#pragma unroll
      for (int v = 0; v < 8; ++v) o[v] = acc[v];
    }
  }
}

// ---------------------------------------------------------------------------
extern "C" void kernel_launch(void* const* d_in, const int* in_sizes, int n_in,
                              void* d_out, int out_size, void* d_ws, size_t ws_size,
                              hipStream_t stream) {
  (void)in_sizes; (void)n_in; (void)out_size; (void)ws_size;
  const float* xyz      = (const float*)d_in[0];
  const float* new_xyz  = (const float*)d_in[1];
  const float* rot      = (const float*)d_in[2];
  const float* features = (const float*)d_in[3];
  float* out    = (float*)d_out;
  int*   ws_idx = (int*)d_ws;              // B*P*S ints = 512 KB of scratch

  // dynamic LDS: score[N] + 2 staging tiles + rv[256] + ri[256] + idx[64] + misc
  const size_t smem = (size_t)N_ * 4 + 2 * TILE_BYTES_ + 512 * 4 + S_ * 4 + 16;

  cyl_query_kernel<<<dim3(B_ * P_), dim3(256), smem, stream>>>(
      xyz, new_xyz, rot, ws_idx);
  group_feat_kernel<<<dim3(B_ * P_, C_ / 4), dim3(256), 0, stream>>>(
      features, ws_idx, out);
  rotate_xyz_wmma_kernel<<<dim3(B_ * P_), dim3(32), 0, stream>>>(
      xyz, new_xyz, rot, ws_idx, out);
}